// Group_Attention_86887188398248
// MI455X (gfx1250) — compile-verified
//
#include <hip/hip_runtime.h>

// ---------------------------------------------------------------------------
// Problem constants (B=2, C=256, N=4096), fp32 end-to-end.
// ---------------------------------------------------------------------------
constexpr int kB = 2;
constexpr int kC = 256;
constexpr int kN = 4096;
constexpr float kEps = 1e-5f;
constexpr float kInvT = 0.0625f;            // 1/sqrt(C) = 1/16
constexpr int kBCN = kB * kC * kN;          // 2,097,152 floats per tensor
constexpr int kRowTiles = kN / 16;          // 256

typedef float v2f __attribute__((ext_vector_type(2)));
typedef float v8f __attribute__((ext_vector_type(8)));

// V_WMMA_F32_16X16X4_F32: A=16x4 f32 (2 VGPR), B=4x16 f32 (2 VGPR), C/D 8 VGPR
__device__ __forceinline__ v8f wmma4(v2f a, v2f b, v8f c) {
  return __builtin_amdgcn_wmma_f32_16x16x4_f32(false, a, false, b, (short)0, c,
                                               false, false);
}

// ---------------------------------------------------------------------------
// Tensor Data Mover (TDM): wave-level 2D tiled DMA global->LDS, tracked with
// TENSORcnt. Descriptor packing per CDNA5 ISA ch.8 (D# groups 0/1).
// This toolchain exposes the 6-arg builtin:
//   (u32x4 g0, i32x8 g1, i32x4 g2, i32x4 g3, i32x8 extra, i32 cpol)
// ---------------------------------------------------------------------------
#if __has_builtin(__builtin_amdgcn_tensor_load_to_lds) && \
    __has_builtin(__builtin_amdgcn_s_wait_tensorcnt)
#define HAVE_TDM 1
typedef unsigned int u32x4 __attribute__((ext_vector_type(4)));
typedef int i32x4v __attribute__((ext_vector_type(4)));
typedef int i32x8v __attribute__((ext_vector_type(8)));

__device__ __forceinline__ unsigned lds_off(const void* p) {
  return (unsigned)(size_t)p;  // flat LDS aperture: addr[31:0] = LDS byte offset
}

// 2D tile load: tile_d0 = contiguous elements per line, tile_d1 = lines,
// stride0 = elements between lines in memory. Optional LDS padding:
// pad_interval code 7 = every 256 dwords, pad_amount code 3 = +4 dwords.
__device__ __forceinline__ void tdm_load_2d(const void* gptr, unsigned ldsb,
                                            unsigned tile_d0, unsigned tile_d1,
                                            unsigned tensor_d0, unsigned tensor_d1,
                                            unsigned stride0, unsigned pad_int,
                                            unsigned pad_amt, bool pad_en) {
  unsigned long long ga = (unsigned long long)(size_t)gptr;
  u32x4 g0 = {1u,                                   // count=1, user descriptor
              ldsb,                                 // lds_addr
              (unsigned)(ga & 0xFFFFFFFFu),         // global_addr[31:0]
              (unsigned)((ga >> 32) & 0x01FFFFFFu)  // global_addr[56:32]
                  | 0x80000000u};                   // type=2 ("image")
  unsigned w0 = (2u << 16)                          // data_size = 4B
                | (pad_en ? (1u << 20) : 0u) | (pad_int << 22) | (pad_amt << 25);
  i32x8v g1 = {(int)w0,
               (int)((tensor_d0 & 0xFFFFu) << 16),
               (int)(((tensor_d0 >> 16) & 0xFFFFu) | ((tensor_d1 & 0xFFFFu) << 16)),
               (int)(((tensor_d1 >> 16) & 0xFFFFu) | ((tile_d0 & 0xFFFFu) << 16)),
               (int)(tile_d1 & 0xFFFFu),            // tile_dim1, tile_dim2=0
               (int)stride0, 0, 0};                 // 48b dim0 stride, dim1 stride=0
  i32x4v z4 = {0, 0, 0, 0};
  i32x8v z8 = {0, 0, 0, 0, 0, 0, 0, 0};
  __builtin_amdgcn_tensor_load_to_lds(g0, g1, z4, z4, z8, 0);
}
#else
#define HAVE_TDM 0
#endif

// Half-wave (16-lane) reductions: one output row of a 16x16 D tile lives in
// one 16-lane half of one VGPR (wave32).
__device__ __forceinline__ float hmax16(float v) {
#pragma unroll
  for (int d = 1; d < 16; d <<= 1) v = fmaxf(v, __shfl_xor(v, d, 16));
  return v;
}
__device__ __forceinline__ float hsum16(float v) {
#pragma unroll
  for (int d = 1; d < 16; d <<= 1) v += __shfl_xor(v, d, 16);
  return v;
}

// ---------------------------------------------------------------------------
__global__ void k_zero(float* p, int n) {
  int i = blockIdx.x * blockDim.x + threadIdx.x;
  if (i < n) p[i] = 0.f;
}

// ---------------------------------------------------------------------------
// feat = linear_w @ x  (per batch: 256x256 @ 256x4096, WMMA f32 16x16x4)
// W tile + X slab staged in LDS via TDM; writes q (C,N), qT (N,C), vT=elu^T.
// One wave per 16x16 output tile.
// ---------------------------------------------------------------------------
__global__ void k_linear(const float* __restrict__ W, const float* __restrict__ X,
                         float* __restrict__ q, float* __restrict__ qT,
                         float* __restrict__ vT) {
  __shared__ float sW[16 * 260];  // rows m0..m0+15 of W, padded stride
  __shared__ float sX[256 * 16];  // cols n0..n0+15 of X, row-major [k][n]
  const int lane = threadIdx.x & 31, half = lane >> 4, l16 = lane & 15;
  const int n0 = blockIdx.x * 16, m0 = blockIdx.y * 16, b = blockIdx.z;
  const float* Xb = X + (size_t)b * kC * kN;
#if HAVE_TDM
  tdm_load_2d(W + (size_t)m0 * kC, lds_off(sW), 256, 16, 256, 256, 256, 7, 3, true);
  tdm_load_2d(Xb + n0, lds_off(sX), 16, 256, 4096, 256, 4096, 0, 0, false);
  __builtin_amdgcn_s_wait_tensorcnt(0);
#else
  for (int i = lane; i < 16 * 256; i += 32)
    sW[(i >> 8) * 260 + (i & 255)] = W[(size_t)(m0 + (i >> 8)) * kC + (i & 255)];
  for (int i = lane; i < 256 * 16; i += 32)
    sX[i] = Xb[(size_t)(i >> 4) * kN + n0 + (i & 15)];
#endif
  __syncthreads();
  v8f acc = {};
  for (int k0 = 0; k0 < kC; k0 += 4) {
    v2f a = *(const v2f*)&sW[l16 * 260 + k0 + 2 * half];
    v2f bb = {sX[(k0 + 2 * half) * 16 + l16], sX[(k0 + 1 + 2 * half) * 16 + l16]};
    acc = wmma4(a, bb, acc);
  }
#pragma unroll
  for (int j = 0; j < 8; ++j) {
    const int row = m0 + j + 8 * half;
    const int col = n0 + l16;
    const float f = acc[j];
    q[(size_t)b * kC * kN + (size_t)row * kN + col] = f;
    qT[(size_t)b * kN * kC + (size_t)col * kC + row] = f;
    vT[(size_t)b * kN * kC + (size_t)col * kC + row] =
        f > 0.f ? f : (__expf(f) - 1.f);  // elu(alpha=1)
  }
}

// ---------------------------------------------------------------------------
// Fused masked attention, flash-style (never materializes the NxN matrix).
// One WG (8 waves) per (batch, 16 query rows). Waves partition the 256 key
// tiles with online softmax; combine across waves via LDS; emits per-WG
// partial sums for the InstanceNorm over val.
// ---------------------------------------------------------------------------
__global__ void __launch_bounds__(256) k_attn(
    const float* __restrict__ qg, const float* __restrict__ qT,
    const float* __restrict__ vT, const int* __restrict__ em,
    const float* __restrict__ ea, float* __restrict__ val,
    float* __restrict__ vpart) {
  __shared__ float sQ[16 * 260];     // query tile (N,C), TDM-padded stride
  __shared__ float sO[16 * 260];     // cross-wave output accumulator
  __shared__ float sP[8][16][18];    // per-wave P tile staging (D->A transpose)
  __shared__ float sM[8][16];
  __shared__ float sL[8][16];
  __shared__ float gL[16];
  __shared__ float red1[8], red2[8];

  const int tid = threadIdx.x;
  const int w = tid >> 5, lane = tid & 31, half = lane >> 4, l16 = lane & 15;
  const int b = blockIdx.y, n0 = blockIdx.x * 16;
  const float NEGINF = -__builtin_inff();

#if HAVE_TDM
  if (w == 0) {  // wave-uniform: TDM issues once per wave
    tdm_load_2d(qT + (size_t)b * kN * kC + (size_t)n0 * kC, lds_off(sQ),
                256, 16, 256, 4096, 256, 7, 3, true);
    __builtin_amdgcn_s_wait_tensorcnt(0);
  }
#else
  for (int i = tid; i < 16 * 256; i += 256)
    sQ[(i >> 8) * 260 + (i & 255)] =
        qT[(size_t)b * kN * kC + (size_t)(n0 + (i >> 8)) * kC + (i & 255)];
#endif
  for (int i = tid; i < 16 * 260; i += 256) sO[i] = 0.f;
  __syncthreads();

  const float* Qb = qg + (size_t)b * kC * kN;   // (C,N) for S B-frags
  const float* Vb = vT + (size_t)b * kN * kC;   // (N,C) for PV B-frags
  const float* EA = ea + (size_t)b * (size_t)kN * kN;
  const int* EM = em + (size_t)b * (size_t)kN * kN;

  float m_run[8], l_run[8];
  v8f o[16];
#pragma unroll
  for (int j = 0; j < 8; ++j) { m_run[j] = NEGINF; l_run[j] = 0.f; }
#pragma unroll
  for (int cc = 0; cc < 16; ++cc) { v8f z = {}; o[cc] = z; }

  for (int ct = w; ct < kRowTiles; ct += 8) {
    const int m0 = ct * 16;
    __builtin_prefetch(EA + (size_t)(n0 + l16) * kN + m0 + 128, 0, 1);
    __builtin_prefetch(EM + (size_t)(n0 + l16) * kN + m0 + 128, 0, 1);

    // ---- S = Q_tile @ K_tile^T  (K = C = 256 -> 64 wmma) ----
    v8f s = {};
    for (int kk = 0; kk < 64; ++kk) {
      v2f a = *(const v2f*)&sQ[l16 * 260 + kk * 4 + 2 * half];
      v2f bb = {Qb[(size_t)(kk * 4 + 2 * half) * kN + m0 + l16],
                Qb[(size_t)(kk * 4 + 1 + 2 * half) * kN + m0 + l16]};
      s = wmma4(a, bb, s);
    }

    // ---- masked, edge-scaled online softmax ----
    float scale[8];
#pragma unroll
    for (int j = 0; j < 8; ++j) {
      const int r = n0 + j + 8 * half;     // query row (D layout)
      const int mc = m0 + l16;             // key column
      const int msk = EM[(size_t)r * kN + mc];
      const float eaw = EA[(size_t)r * kN + mc];
      const float logit = msk ? eaw * s[j] * kInvT : NEGINF;
      const float mnew = fmaxf(m_run[j], hmax16(logit));
      const float p = msk ? __expf(logit - mnew) : 0.f;
      const float rs = hsum16(p);
      const float sc = (m_run[j] > NEGINF) ? __expf(m_run[j] - mnew) : 0.f;
      l_run[j] = l_run[j] * sc + rs;
      m_run[j] = mnew;
      scale[j] = sc;
      sP[w][j + 8 * half][l16] = p;        // stage P (same-wave LDS, in order)
    }
#pragma unroll
    for (int cc = 0; cc < 16; ++cc)
#pragma unroll
      for (int j = 0; j < 8; ++j) o[cc][j] *= scale[j];

    // ---- O += P @ V_tile  (K = 16 -> 4 wmma per 16-channel tile) ----
    for (int cc = 0; cc < 16; ++cc) {
      const int c0 = cc * 16;
#pragma unroll
      for (int kk = 0; kk < 4; ++kk) {
        v2f a = *(const v2f*)&sP[w][l16][kk * 4 + 2 * half];
        const int kr = m0 + kk * 4 + 2 * half;
        v2f bb = {Vb[(size_t)kr * kC + c0 + l16],
                  Vb[(size_t)(kr + 1) * kC + c0 + l16]};
        o[cc] = wmma4(a, bb, o[cc]);
      }
    }
  }

  // ---- cross-wave combine ----
  if (l16 == 0) {
#pragma unroll
    for (int j = 0; j < 8; ++j) {
      sM[w][j + 8 * half] = m_run[j];
      sL[w][j + 8 * half] = l_run[j];
    }
  }
  __syncthreads();

  float fac[8];
#pragma unroll
  for (int j = 0; j < 8; ++j) {
    const int r = j + 8 * half;
    float gm = NEGINF;
    for (int w2 = 0; w2 < 8; ++w2) gm = fmaxf(gm, sM[w2][r]);
    float gl = 0.f;
    for (int w2 = 0; w2 < 8; ++w2) {
      const float mw = sM[w2][r];
      gl += (mw > NEGINF) ? sL[w2][r] * __expf(mw - gm) : 0.f;
    }
    fac[j] = (m_run[j] > NEGINF) ? __expf(m_run[j] - gm) : 0.f;
    if (w == 0 && l16 == 0) gL[r] = gl;
  }
#pragma unroll
  for (int cc = 0; cc < 16; ++cc)
#pragma unroll
    for (int j = 0; j < 8; ++j)
      atomicAdd(&sO[(j + 8 * half) * 260 + cc * 16 + l16], o[cc][j] * fac[j]);
  __syncthreads();

  // ---- write val (B,N,C) + partial InstanceNorm stats ----
  float s1 = 0.f, s2 = 0.f;
#pragma unroll
  for (int i = 0; i < 16; ++i) {
    const int idx = tid + 256 * i;
    const int row = idx >> 8, c = idx & 255;
    const float vv = sO[row * 260 + c] / gL[row];
    val[(size_t)b * kN * kC + (size_t)(n0 + row) * kC + c] = vv;
    s1 += vv;
    s2 += vv * vv;
  }
#pragma unroll
  for (int d = 1; d < 32; d <<= 1) {
    s1 += __shfl_xor(s1, d, 32);
    s2 += __shfl_xor(s2, d, 32);
  }
  if (lane == 0) { red1[w] = s1; red2[w] = s2; }
  __syncthreads();
  if (tid == 0) {
    float a1 = 0.f, a2 = 0.f;
    for (int i = 0; i < 8; ++i) { a1 += red1[i]; a2 += red2[i]; }
    vpart[((size_t)b * kRowTiles + blockIdx.x) * 2 + 0] = a1;
    vpart[((size_t)b * kRowTiles + blockIdx.x) * 2 + 1] = a2;
  }
}

// Reduce per-WG partials -> per-batch mean/rstd (deterministic, no atomics).
__global__ void k_valfin(const float* __restrict__ vpart, float* __restrict__ mv) {
  const int b = threadIdx.x;
  if (b < kB) {
    float s1 = 0.f, s2 = 0.f;
    for (int i = 0; i < kRowTiles; ++i) {
      s1 += vpart[((size_t)b * kRowTiles + i) * 2 + 0];
      s2 += vpart[((size_t)b * kRowTiles + i) * 2 + 1];
    }
    const float inv = 1.f / (float)(kN * kC);
    const float m = s1 * inv;
    const float var = s2 * inv - m * m;
    mv[b * 2 + 0] = m;
    mv[b * 2 + 1] = rsqrtf(var + kEps);
  }
}

// feat_attn(B,C,N) = IN(val)^T + x, via LDS 16x16 transpose (coalesced both ways)
__global__ void k_fa(const float* __restrict__ val, const float* __restrict__ mv,
                     const float* __restrict__ x, float* __restrict__ fa) {
  __shared__ float tile[16][17];
  const int tx = threadIdx.x, ty = threadIdx.y;
  const int n0 = blockIdx.x * 16, c0 = blockIdx.y * 16, b = blockIdx.z;
  tile[ty][tx] = val[(size_t)b * kN * kC + (size_t)(n0 + ty) * kC + c0 + tx];
  __syncthreads();
  const float m = mv[b * 2], r = mv[b * 2 + 1];
  const size_t xi = (size_t)b * kC * kN + (size_t)(c0 + ty) * kN + n0 + tx;
  fa[xi] = (tile[tx][ty] - m) * r + x[xi];
}

// ---------------------------------------------------------------------------
// Generic conv1x1 GEMM (WMMA f32, TDM-staged operands) with fused stat
// accumulation. statMode: 1 = per-channel (BN feed), 2 = per-(b,c) (IN feed)
// ---------------------------------------------------------------------------
__global__ void k_conv(const float* __restrict__ W, const float* __restrict__ bias,
                       const float* __restrict__ in, float* __restrict__ out,
                       float* s1, float* s2, int statMode) {
  __shared__ float sW[16 * 260];
  __shared__ float sX[256 * 16];
  const int lane = threadIdx.x & 31, half = lane >> 4, l16 = lane & 15;
  const int n0 = blockIdx.x * 16, m0 = blockIdx.y * 16, b = blockIdx.z;
  const float* Xb = in + (size_t)b * kC * kN;
#if HAVE_TDM
  tdm_load_2d(W + (size_t)m0 * kC, lds_off(sW), 256, 16, 256, 256, 256, 7, 3, true);
  tdm_load_2d(Xb + n0, lds_off(sX), 16, 256, 4096, 256, 4096, 0, 0, false);
  __builtin_amdgcn_s_wait_tensorcnt(0);
#else
  for (int i = lane; i < 16 * 256; i += 32)
    sW[(i >> 8) * 260 + (i & 255)] = W[(size_t)(m0 + (i >> 8)) * kC + (i & 255)];
  for (int i = lane; i < 256 * 16; i += 32)
    sX[i] = Xb[(size_t)(i >> 4) * kN + n0 + (i & 15)];
#endif
  __syncthreads();
  v8f acc = {};
  for (int k0 = 0; k0 < kC; k0 += 4) {
    v2f a = *(const v2f*)&sW[l16 * 260 + k0 + 2 * half];
    v2f bb = {sX[(k0 + 2 * half) * 16 + l16], sX[(k0 + 1 + 2 * half) * 16 + l16]};
    acc = wmma4(a, bb, acc);
  }
#pragma unroll
  for (int j = 0; j < 8; ++j) {
    const int row = m0 + j + 8 * half;
    const float f = acc[j] + bias[row];
    out[(size_t)b * kC * kN + (size_t)row * kN + n0 + l16] = f;
    if (statMode) {
      const float r1 = hsum16(f);
      const float r2 = hsum16(f * f);
      if (l16 == 0) {
        const int idx = (statMode == 2) ? b * kC + row : row;
        atomicAdd(&s1[idx], r1);
        atomicAdd(&s2[idx], r2);
      }
    }
  }
}

// BN(IN(y)) collapses to per-(b,c) affine: BN mean of IN output is exactly 0,
// BN var_c = mean_b[v_bc/(v_bc+eps)]. alpha = r_bc*bnr_c*w[c], beta = b[c]-m*alpha.
__global__ void k_scales(const float* __restrict__ s1, const float* __restrict__ s2,
                         const float* __restrict__ bw, const float* __restrict__ bb,
                         float* __restrict__ ab) {
  const int c = threadIdx.x;
  if (c < kC) {
    float mb[kB], rb[kB];
    float vc = 0.f;
    for (int b = 0; b < kB; ++b) {
      const float m = s1[b * kC + c] / (float)kN;
      const float v = s2[b * kC + c] / (float)kN - m * m;
      mb[b] = m;
      const float r = rsqrtf(v + kEps);
      rb[b] = r;
      vc += v * r * r;  // v/(v+eps)
    }
    vc *= (1.f / kB);
    const float bnr = rsqrtf(vc + kEps);
    for (int b = 0; b < kB; ++b) {
      const float al = rb[b] * bnr * bw[c];
      ab[(b * kC + c) * 2 + 0] = al;
      ab[(b * kC + c) * 2 + 1] = bb[c] - mb[b] * al;
    }
  }
}

// Plain BatchNorm scales (right branch): gamma/delta per channel
__global__ void k_bnr(const float* __restrict__ s1, const float* __restrict__ s2,
                      const float* __restrict__ bw, const float* __restrict__ bb,
                      float* __restrict__ gd) {
  const int c = threadIdx.x;
  if (c < kC) {
    const float inv = 1.f / (float)(kB * kN);
    const float m = s1[c] * inv;
    const float var = s2[c] * inv - m * m;
    const float g = bw[c] * rsqrtf(var + kEps);
    gd[c * 2 + 0] = g;
    gd[c * 2 + 1] = bb[c] - m * g;
  }
}

// h = relu(alpha*t1 + beta) in place
__global__ void k_act(float* t, const float* __restrict__ ab) {
  const int i = blockIdx.x * blockDim.x + threadIdx.x;
  if (i < kBCN) {
    const int bc = i / kN;  // b*kC + c
    const float v = ab[bc * 2] * t[i] + ab[bc * 2 + 1];
    t[i] = v > 0.f ? v : 0.f;
  }
}

// out = relu( (a2*t2 + b2) + (gamma*tr + delta) )
__global__ void k_final(const float* __restrict__ t2, const float* __restrict__ ab2,
                        const float* __restrict__ tr, const float* __restrict__ gd,
                        float* __restrict__ out) {
  const int i = blockIdx.x * blockDim.x + threadIdx.x;
  if (i < kBCN) {
    const int bc = i / kN;
    const int c = bc & (kC - 1);
    const float y2 = ab2[bc * 2] * t2[i] + ab2[bc * 2 + 1];
    const float x1 = gd[c * 2] * tr[i] + gd[c * 2 + 1];
    const float r = y2 + x1;
    out[i] = r > 0.f ? r : 0.f;
  }
}

// ---------------------------------------------------------------------------
extern "C" void kernel_launch(void* const* d_in, const int* in_sizes, int n_in,
                              void* d_out, int out_size, void* d_ws, size_t ws_size,
                              hipStream_t stream) {
  (void)in_sizes; (void)n_in; (void)out_size; (void)ws_size;
  const float* x    = (const float*)d_in[0];
  const int*   em   = (const int*)d_in[1];
  const float* ea   = (const float*)d_in[2];
  const float* lw   = (const float*)d_in[3];
  const float* wr   = (const float*)d_in[4];
  const float* br   = (const float*)d_in[5];
  const float* bnrw = (const float*)d_in[6];
  const float* bnrb = (const float*)d_in[7];
  const float* w1   = (const float*)d_in[8];
  const float* b1   = (const float*)d_in[9];
  const float* bn1w = (const float*)d_in[10];
  const float* bn1b = (const float*)d_in[11];
  const float* w2   = (const float*)d_in[12];
  const float* b2   = (const float*)d_in[13];
  const float* bn2w = (const float*)d_in[14];
  const float* bn2b = (const float*)d_in[15];
  float* out = (float*)d_out;

  // workspace carve (7 x 8MiB tensors + ~24KB stats; t2 aliases q)
  float* ws  = (float*)d_ws;
  float* q   = ws;
  float* qT  = q + kBCN;
  float* vT  = qT + kBCN;
  float* val = vT + kBCN;
  float* fa  = val + kBCN;
  float* tr  = fa + kBCN;
  float* t1  = tr + kBCN;
  float* t2  = q;  // q is dead after attention
  float* st    = t1 + kBCN;
  float* vpart = st;                           // kB*kRowTiles*2
  float* mv    = vpart + kB * kRowTiles * 2;   // kB*2
  float* bnr_s = mv + kB * 2;                  // kC
  float* bnr_q = bnr_s + kC;                   // kC
  float* in1_s = bnr_q + kC;                   // kB*kC
  float* in1_q = in1_s + kB * kC;
  float* in2_s = in1_q + kB * kC;
  float* in2_q = in2_s + kB * kC;
  float* ab1   = in2_q + kB * kC;              // kB*kC*2
  float* ab2   = ab1 + kB * kC * 2;
  float* gd    = ab2 + kB * kC * 2;            // kC*2
  const int nstats = (int)((gd + kC * 2) - st);

  k_zero<<<(nstats + 255) / 256, 256, 0, stream>>>(st, nstats);
  k_linear<<<dim3(kN / 16, kC / 16, kB), 32, 0, stream>>>(lw, x, q, qT, vT);
  k_attn<<<dim3(kRowTiles, kB), 256, 0, stream>>>(q, qT, vT, em, ea, val, vpart);
  k_valfin<<<1, 32, 0, stream>>>(vpart, mv);
  k_fa<<<dim3(kN / 16, kC / 16, kB), dim3(16, 16), 0, stream>>>(val, mv, x, fa);
  k_conv<<<dim3(kN / 16, kC / 16, kB), 32, 0, stream>>>(wr, br, fa, tr, bnr_s, bnr_q, 1);
  k_conv<<<dim3(kN / 16, kC / 16, kB), 32, 0, stream>>>(w1, b1, fa, t1, in1_s, in1_q, 2);
  k_scales<<<1, 256, 0, stream>>>(in1_s, in1_q, bn1w, bn1b, ab1);
  k_act<<<kBCN / 256, 256, 0, stream>>>(t1, ab1);
  k_conv<<<dim3(kN / 16, kC / 16, kB), 32, 0, stream>>>(w2, b2, t1, t2, in2_s, in2_q, 2);
  k_scales<<<1, 256, 0, stream>>>(in2_s, in2_q, bn2w, bn2b, ab2);
  k_bnr<<<1, 256, 0, stream>>>(bnr_s, bnr_q, bnrw, bnrb, gd);
  k_final<<<kBCN / 256, 256, 0, stream>>>(t2, ab2, tr, gd, out);
}